// EFGPND_73400991089335
// MI455X (gfx1250) — compile-verified
//
#include <hip/hip_runtime.h>
#include <math.h>

// ---------------------------------------------------------------------------
// EFGP GP regression on gfx1250 (MI455X).
//  Stage A : Fy (64x64) and v (127x127) complex Gram sums as f32 WMMA GEMMs
//            (V_WMMA_F32_16X16X4_F32), K = N_train, phases on the fly.
//            2x2 super-tiles per block: each sincos feeds two tile rows/cols,
//            16 WMMAs per 4-point K-chunk.
//  Stage B : CG (50 iters) with direct block-Toeplitz matvec (VALU + LDS),
//            fixed-order partial reductions (deterministic, no atomics).
//  Stage C : posterior mean at test points via complex-rotation recurrences.
// ---------------------------------------------------------------------------

#define PI_F 3.14159265358979323846f

typedef __attribute__((ext_vector_type(2))) float v2f;
typedef __attribute__((ext_vector_type(8))) float v8f;

#define M_G   64
#define LDV   128      // padded row stride for the 127x127 v grid
#define NSLAB 32       // K-dimension slabs in stage A

// ---- workspace layout (float offsets) -------------------------------------
#define OFF_VRE    0
#define OFF_VIM    (OFF_VRE + LDV*LDV)
#define OFF_FRE    (OFF_VIM + LDV*LDV)
#define OFF_FIM    (OFF_FRE + M_G*M_G)
#define OFF_XRE    (OFF_FIM + M_G*M_G)
#define OFF_XIM    (OFF_XRE + 4096)
#define OFF_RRE    (OFF_XIM + 4096)
#define OFF_RIM    (OFF_RRE + 4096)
#define OFF_PRE    (OFF_RIM + 4096)
#define OFF_PIM    (OFF_PRE + 4096)
#define OFF_APRE   (OFF_PIM + 4096)
#define OFF_APIM   (OFF_APRE + 4096)
#define OFF_DENOMP (OFF_APIM + 4096)            // 50*16
#define OFF_RZP    (OFF_DENOMP + 50*16)         // 51*16
#define OFF_PARTV  (OFF_RZP + 51*16)            // 64*NSLAB*512
#define OFF_PARTF  (OFF_PARTV + 64*NSLAB*512)   // 16*NSLAB*512
// total ~= 1,386,064 floats ~= 5.6 MB

// ---------------------------------------------------------------------------
// Stage A: one (2x2 super-tile, K-slab) per block. 8 waves split the slab.
// Sub-tile [a-range, b-range]: Re += C*C' + (-S)*S' ; Im += C*S' + S*C'.
// A frag: lane(l): M = l&15, K = 2*(l>>4)+j (VGPR j). B mirrored on N.
// ---------------------------------------------------------------------------
__global__ __launch_bounds__(256) void kPhaseGemm(
    const float* __restrict__ x, const float* __restrict__ y,
    float* __restrict__ part, int tilesPerRow, int aShift,
    int useY, float imSign, int N, int ptsPerBlock)
{
    __shared__ float red[8][512];
    const int tid  = threadIdx.x;
    const int lane = tid & 31;
    const int wave = tid >> 5;          // wave32
    const int m    = lane & 15;
    const int h    = lane >> 4;
    const int superPerRow = tilesPerRow >> 1;
    const int Ta   = blockIdx.x / superPerRow;
    const int Tb   = blockIdx.x % superPerRow;
    const int slab = blockIdx.y;
    const float fa0 = (float)(Ta * 32 + m - aShift);
    const float fa1 = fa0 + 16.0f;
    const float fb0 = (float)(Tb * 32 + m - aShift);
    const float fb1 = fb0 + 16.0f;

    const int ptsPerWave = ptsPerBlock >> 3;
    const int chunks     = ptsPerWave >> 2;
    int n0 = slab * ptsPerBlock + wave * ptsPerWave;

    v8f accRe[2][2];
    v8f accIm[2][2];
#pragma unroll
    for (int ia = 0; ia < 2; ++ia)
#pragma unroll
        for (int ib = 0; ib < 2; ++ib) { accRe[ia][ib] = {}; accIm[ia][ib] = {}; }

    for (int it = 0; it < chunks; ++it, n0 += 4) {
        float ca[2][2], sa[2][2], cb[2][2], sb[2][2];   // [tile][j]
#pragma unroll
        for (int j = 0; j < 2; ++j) {
            int   n   = n0 + 2 * h + j;
            int   ncl = (n < N) ? n : (N - 1);
            float x0  = x[2 * ncl];
            float x1  = x[2 * ncl + 1];
            float w   = (n < N) ? 1.0f : 0.0f;      // zero padded points
            float yv  = useY ? y[ncl] : 1.0f;
            float wy  = w * yv;
            float s, c;
            __sincosf(PI_F * x0 * fa0, &s, &c);
            ca[0][j] = c * wy;  sa[0][j] = s * wy;
            __sincosf(PI_F * x0 * fa1, &s, &c);
            ca[1][j] = c * wy;  sa[1][j] = s * wy;
            __sincosf(PI_F * x1 * fb0, &s, &c);
            cb[0][j] = c * w;   sb[0][j] = s * w;
            __sincosf(PI_F * x1 * fb1, &s, &c);
            cb[1][j] = c * w;   sb[1][j] = s * w;
        }
        v2f Ac[2], As[2], Ans[2], Bc[2], Bs[2];
#pragma unroll
        for (int t = 0; t < 2; ++t) {
            Ac[t]  = {  ca[t][0],  ca[t][1] };
            As[t]  = {  sa[t][0],  sa[t][1] };
            Ans[t] = { -sa[t][0], -sa[t][1] };
            Bc[t]  = {  cb[t][0],  cb[t][1] };
            Bs[t]  = {  sb[t][0],  sb[t][1] };
        }
#pragma unroll
        for (int ia = 0; ia < 2; ++ia)
#pragma unroll
            for (int ib = 0; ib < 2; ++ib) {
                accRe[ia][ib] = __builtin_amdgcn_wmma_f32_16x16x4_f32(
                    false, Ac[ia],  false, Bc[ib], (short)0, accRe[ia][ib], false, false);
                accRe[ia][ib] = __builtin_amdgcn_wmma_f32_16x16x4_f32(
                    false, Ans[ia], false, Bs[ib], (short)0, accRe[ia][ib], false, false);
                accIm[ia][ib] = __builtin_amdgcn_wmma_f32_16x16x4_f32(
                    false, Ac[ia],  false, Bs[ib], (short)0, accIm[ia][ib], false, false);
                accIm[ia][ib] = __builtin_amdgcn_wmma_f32_16x16x4_f32(
                    false, As[ia],  false, Bc[ib], (short)0, accIm[ia][ib], false, false);
            }
    }

    // C/D layout: VGPR r -> M = r + 8*h, N = m.  One reduction pass per
    // sub-tile so LDS stays at 16 KB.
#pragma unroll
    for (int st = 0; st < 4; ++st) {
        const int ia = st >> 1, ib = st & 1;
        __syncthreads();
#pragma unroll
        for (int r = 0; r < 8; ++r) {
            int idx = (r + 8 * h) * 16 + m;
            red[wave][idx]       = accRe[ia][ib][r];
            red[wave][256 + idx] = imSign * accIm[ia][ib][r];
        }
        __syncthreads();
        int tile = (2 * Ta + ia) * tilesPerRow + (2 * Tb + ib);
        for (int e = tid; e < 512; e += 256) {
            float sum = 0.f;
#pragma unroll
            for (int w = 0; w < 8; ++w) sum += red[w][e];   // fixed order
            part[((size_t)tile * NSLAB + slab) * 512 + e] = sum;
        }
    }
}

// Sum K-slab partials into the final (re, im) grids.
__global__ __launch_bounds__(512) void kReduceA(
    const float* __restrict__ part, float* __restrict__ outRe,
    float* __restrict__ outIm, int tilesPerRow, int dim, int ldOut)
{
    int tile = blockIdx.x;
    int e    = threadIdx.x;          // 0..511
    float sum = 0.f;
    const float* p = part + (size_t)tile * NSLAB * 512 + e;
#pragma unroll
    for (int s = 0; s < NSLAB; ++s) sum += p[(size_t)s * 512];
    int isIm = e >> 8;
    int eL   = e & 255;
    int Mr   = eL >> 4, Nn = eL & 15;
    int a = (tile / tilesPerRow) * 16 + Mr;
    int b = (tile % tilesPerRow) * 16 + Nn;
    if (a < dim && b < dim)
        (isIm ? outIm : outRe)[a * ldOut + b] = sum;
}

// r = p = ws*Fy ; x = 0 ; rz0 partials
__global__ __launch_bounds__(256) void kInit(const float* __restrict__ wsq,
                                             float* __restrict__ W)
{
    __shared__ float red[256];
    int tid = threadIdx.x;
    int k   = blockIdx.x * 256 + tid;
    float w  = wsq[k];
    float rr = w * W[OFF_FRE + k];
    float ri = w * W[OFF_FIM + k];
    W[OFF_RRE + k] = rr; W[OFF_RIM + k] = ri;
    W[OFF_PRE + k] = rr; W[OFF_PIM + k] = ri;
    W[OFF_XRE + k] = 0.f; W[OFF_XIM + k] = 0.f;
    red[tid] = rr * rr + ri * ri;
    __syncthreads();
    for (int off = 128; off > 0; off >>= 1) {
        if (tid < off) red[tid] += red[tid + off];
        __syncthreads();
    }
    if (tid == 0) W[OFF_RZP + blockIdx.x] = red[0];
}

// Ap = ws * T(ws*p) + sigma2 * p ; denom partials. t = ws*p staged in LDS.
__global__ __launch_bounds__(256) void kMatvec(const float* __restrict__ wsq,
    const float* __restrict__ sigma2, float* __restrict__ W, int iter)
{
    __shared__ float tRe[4096], tIm[4096];
    __shared__ float red[256];
    int tid = threadIdx.x;
    for (int i = tid; i < 4096; i += 256) {
        float w = wsq[i];
        tRe[i] = w * W[OFF_PRE + i];
        tIm[i] = w * W[OFF_PIM + i];
    }
    __syncthreads();
    int k  = blockIdx.x * 256 + tid;
    int k1 = k >> 6, k2 = k & 63;
    const float* vre = W + OFF_VRE;
    const float* vim = W + OFF_VIM;
    float aRe = 0.f, aIm = 0.f;
    for (int j1 = 0; j1 < 64; ++j1) {
        int base = (k1 - j1 + 63) * LDV + (k2 + 63);
        const float* tr = tRe + j1 * 64;
        const float* ti = tIm + j1 * 64;
        for (int j2 = 0; j2 < 64; ++j2) {
            float vr = vre[base - j2], vi = vim[base - j2];
            float ur = tr[j2], ui = ti[j2];
            aRe += vr * ur - vi * ui;
            aIm += vr * ui + vi * ur;
        }
    }
    float w   = wsq[k];
    float s2  = sigma2[0];
    float pr  = W[OFF_PRE + k], pi = W[OFF_PIM + k];
    float apr = w * aRe + s2 * pr;
    float api = w * aIm + s2 * pi;
    W[OFF_APRE + k] = apr;
    W[OFF_APIM + k] = api;
    red[tid] = pr * apr + pi * api;
    __syncthreads();
    for (int off = 128; off > 0; off >>= 1) {
        if (tid < off) red[tid] += red[tid + off];
        __syncthreads();
    }
    if (tid == 0) W[OFF_DENOMP + iter * 16 + blockIdx.x] = red[0];
}

// x += alpha p ; r -= alpha Ap ; rz_new partials
__global__ __launch_bounds__(256) void kXR(float* __restrict__ W, int iter)
{
    __shared__ float red[256];
    int tid = threadIdx.x;
    int k   = blockIdx.x * 256 + tid;
    float rz = 0.f, dn = 0.f;
#pragma unroll
    for (int b = 0; b < 16; ++b) {
        rz += W[OFF_RZP + iter * 16 + b];
        dn += W[OFF_DENOMP + iter * 16 + b];
    }
    float alpha = rz / (dn + 1e-30f);
    float pr = W[OFF_PRE + k], pi = W[OFF_PIM + k];
    W[OFF_XRE + k] += alpha * pr;
    W[OFF_XIM + k] += alpha * pi;
    float rr = W[OFF_RRE + k] - alpha * W[OFF_APRE + k];
    float ri = W[OFF_RIM + k] - alpha * W[OFF_APIM + k];
    W[OFF_RRE + k] = rr;
    W[OFF_RIM + k] = ri;
    red[tid] = rr * rr + ri * ri;
    __syncthreads();
    for (int off = 128; off > 0; off >>= 1) {
        if (tid < off) red[tid] += red[tid + off];
        __syncthreads();
    }
    if (tid == 0) W[OFF_RZP + (iter + 1) * 16 + blockIdx.x] = red[0];
}

// p = r + beta p
__global__ __launch_bounds__(256) void kP(float* __restrict__ W, int iter)
{
    int k = blockIdx.x * 256 + threadIdx.x;
    float rzOld = 0.f, rzNew = 0.f;
#pragma unroll
    for (int b = 0; b < 16; ++b) {
        rzOld += W[OFF_RZP + iter * 16 + b];
        rzNew += W[OFF_RZP + (iter + 1) * 16 + b];
    }
    float beta = rzNew / (rzOld + 1e-30f);
    W[OFF_PRE + k] = W[OFF_RRE + k] + beta * W[OFF_PRE + k];
    W[OFF_PIM + k] = W[OFF_RIM + k] + beta * W[OFF_PIM + k];
}

// mu_b = Re( sum_{j,k} T1[b,j] * (ws*x)[j,k] * T2[b,k] ), phases by recurrence.
__global__ __launch_bounds__(256) void kPost(const float* __restrict__ xnew,
    const float* __restrict__ wsq, const float* __restrict__ W,
    float* __restrict__ out, int B)
{
    __shared__ float aRe[4096], aIm[4096];
    int tid = threadIdx.x;
    for (int i = tid; i < 4096; i += 256) {
        float w = wsq[i];
        aRe[i] = w * W[OFF_XRE + i];
        aIm[i] = w * W[OFF_XIM + i];
    }
    __syncthreads();
    int b = blockIdx.x * 256 + tid;
    if (b >= B) return;
    float x0 = xnew[2 * b], x1 = xnew[2 * b + 1];
    float c1, s1, c2, s2, c1s, s1s, c2s, s2s;
    __sincosf(PI_F * x0, &s1, &c1);                 // step exp(i*pi*x0)
    __sincosf(PI_F * x1, &s2, &c2);
    __sincosf(-32.0f * PI_F * x0, &s1s, &c1s);      // start exp(-32*i*pi*x0)
    __sincosf(-32.0f * PI_F * x1, &s2s, &c2s);
    float t1r = c1s, t1i = s1s;
    float mu = 0.f;
    for (int j = 0; j < 64; ++j) {
        float t2r = c2s, t2i = s2s;
        float sr = 0.f, si = 0.f;
        const float* ar = aRe + j * 64;
        const float* ai = aIm + j * 64;
        for (int kk = 0; kk < 64; ++kk) {
            float arv = ar[kk], aiv = ai[kk];
            sr += arv * t2r - aiv * t2i;
            si += arv * t2i + aiv * t2r;
            float nr = t2r * c2 - t2i * s2;
            t2i = t2r * s2 + t2i * c2;
            t2r = nr;
        }
        mu += t1r * sr - t1i * si;
        float nr = t1r * c1 - t1i * s1;
        t1i = t1r * s1 + t1i * c1;
        t1r = nr;
    }
    out[b] = mu;
}

extern "C" void kernel_launch(void* const* d_in, const int* in_sizes, int n_in,
                              void* d_out, int out_size, void* d_ws, size_t ws_size,
                              hipStream_t stream)
{
    (void)n_in; (void)out_size; (void)ws_size;
    const float* x      = (const float*)d_in[0];
    const float* y      = (const float*)d_in[1];
    const float* xnew   = (const float*)d_in[2];
    const float* wsq    = (const float*)d_in[3];
    const float* sigma2 = (const float*)d_in[4];
    float* W = (float*)d_ws;

    int N = in_sizes[0] / 2;
    int B = in_sizes[2] / 2;
    int ptsPerBlock = (((N + NSLAB - 1) / NSLAB) + 31) & ~31;  // mult of 32

    // Stage A: v (8x8 tiles -> 4x4 super-tiles) and Fy (4x4 -> 2x2 supers).
    kPhaseGemm<<<dim3(16, NSLAB), 256, 0, stream>>>(x, y, W + OFF_PARTV,
                                                    8, 63, 0,  1.0f, N, ptsPerBlock);
    kPhaseGemm<<<dim3(4, NSLAB), 256, 0, stream>>>(x, y, W + OFF_PARTF,
                                                   4, 32, 1, -1.0f, N, ptsPerBlock);
    kReduceA<<<64, 512, 0, stream>>>(W + OFF_PARTV, W + OFF_VRE, W + OFF_VIM, 8, 127, LDV);
    kReduceA<<<16, 512, 0, stream>>>(W + OFF_PARTF, W + OFF_FRE, W + OFF_FIM, 4, 64, 64);

    // Stage B: CG, 50 fixed iterations.
    kInit<<<16, 256, 0, stream>>>(wsq, W);
    for (int it = 0; it < 50; ++it) {
        kMatvec<<<16, 256, 0, stream>>>(wsq, sigma2, W, it);
        kXR<<<16, 256, 0, stream>>>(W, it);
        kP<<<16, 256, 0, stream>>>(W, it);
    }

    // Stage C: posterior mean.
    kPost<<<(B + 255) / 256, 256, 0, stream>>>(xnew, wsq, W, (float*)d_out, B);
}